// AI4MULTI_32195074850827
// MI455X (gfx1250) — compile-verified
//
#include <hip/hip_runtime.h>
#include <stdint.h>

#ifndef __has_builtin
#define __has_builtin(x) 0
#endif

// ---------------- problem constants ----------------
#define NZ0 64
#define NY0 64
#define NX0 512
#define NCELL (NZ0*NY0*NX0)

static constexpr float DT_C    = 0.002f;
static constexpr float DX_C    = 0.04f;
static constexpr float GZ_C    = -10.0f;
static constexpr float RHOL_C  = 1000.0f;
static constexpr float RHOG_C  = 1.0f;
static constexpr float NU_C    = 1.0e-3f;
static constexpr float INV_DX2 = 1.0f/(DX_C*DX_C);   // 625
static constexpr float INV2DX  = 1.0f/(2.0f*DX_C);   // 12.5
static constexpr float NU_DX2  = NU_C*INV_DX2;       // 0.625
static constexpr float DIAG_C  = -6.0f*INV_DX2;      // -3750
static constexpr float INVDIAG = 1.0f/DIAG_C;
static constexpr float BSCALE  = -(DX_C*DX_C)/DT_C;  // -0.8

typedef float v2f  __attribute__((ext_vector_type(2)));
typedef float v8f  __attribute__((ext_vector_type(8)));
typedef unsigned int u32x4 __attribute__((ext_vector_type(4)));
typedef int   i32x4 __attribute__((ext_vector_type(4)));
typedef int   i32x8 __attribute__((ext_vector_type(8)));

#if defined(__HIP_DEVICE_COMPILE__) && __has_builtin(__builtin_amdgcn_wmma_f32_16x16x4_f32)
#define USE_WMMA 1
#else
#define USE_WMMA 0
#endif
#if defined(__HIP_DEVICE_COMPILE__) && __has_builtin(__builtin_amdgcn_tensor_load_to_lds)
#define USE_TDM 1
#else
#define USE_TDM 0
#endif

__device__ __forceinline__ int iclamp(int v, int lo, int hi){
  v = v < lo ? lo : v;
  return v > hi ? hi : v;
}

// Branchless ghost-aware load: ALWAYS loads from a clamped (in-range) address,
// then selects 0 for Dirichlet ghosts. No exec-mask divergence.
// Mode per axis: 0 = Neumann (clamp), 1 = Dirichlet (zero),
// 2 = pd-z rule (clamp at bottom, zero above the top face).
template<int MX,int MY,int MZ>
__device__ __forceinline__ float gval(const float* __restrict__ f, int z,int y,int x,
                                      int nz,int ny,int nx){
  bool zero = false;
  if (MX==1) zero = zero || (x < 0) || (x >= nx);
  if (MY==1) zero = zero || (y < 0) || (y >= ny);
  if (MZ==1) zero = zero || (z < 0) || (z >= nz);
  if (MZ==2) zero = zero || (z >= nz);
  x = iclamp(x, 0, nx-1);
  y = iclamp(y, 0, ny-1);
  z = iclamp(z, 0, nz-1);
  float v = f[((size_t)z*ny + y)*nx + x];
  return zero ? 0.0f : v;
}

// ---------------- predictor: u*,v*,w* (fused rho/advect/diffuse/buoyancy) ----------------
__global__ void predictor_kernel(const float* __restrict__ al, const float* __restrict__ u,
                                 const float* __restrict__ v, const float* __restrict__ w,
                                 float* __restrict__ us, float* __restrict__ vs,
                                 float* __restrict__ wst){
  int i = blockIdx.x*blockDim.x + threadIdx.x;
  if (i >= NCELL) return;
  int x = i % NX0; int t = i / NX0; int y = t % NY0; int z = t / NY0;
  float uc = u[i], vc = v[i], wc = w[i];

  float uxm=gval<1,0,0>(u,z,y,x-1,NZ0,NY0,NX0), uxp=gval<1,0,0>(u,z,y,x+1,NZ0,NY0,NX0);
  float uym=gval<1,0,0>(u,z,y-1,x,NZ0,NY0,NX0), uyp=gval<1,0,0>(u,z,y+1,x,NZ0,NY0,NX0);
  float uzm=gval<1,0,0>(u,z-1,y,x,NZ0,NY0,NX0), uzp=gval<1,0,0>(u,z+1,y,x,NZ0,NY0,NX0);
  float rhs_u = (uxm+uxp+uym+uyp+uzm+uzp - 6.0f*uc)*NU_DX2
              - (uc*(uxp-uxm) + vc*(uyp-uym) + wc*(uzp-uzm))*INV2DX;

  float vxm=gval<0,1,0>(v,z,y,x-1,NZ0,NY0,NX0), vxp=gval<0,1,0>(v,z,y,x+1,NZ0,NY0,NX0);
  float vym=gval<0,1,0>(v,z,y-1,x,NZ0,NY0,NX0), vyp=gval<0,1,0>(v,z,y+1,x,NZ0,NY0,NX0);
  float vzm=gval<0,1,0>(v,z-1,y,x,NZ0,NY0,NX0), vzp=gval<0,1,0>(v,z+1,y,x,NZ0,NY0,NX0);
  float rhs_v = (vxm+vxp+vym+vyp+vzm+vzp - 6.0f*vc)*NU_DX2
              - (uc*(vxp-vxm) + vc*(vyp-vym) + wc*(vzp-vzm))*INV2DX;

  float wxm=gval<0,0,1>(w,z,y,x-1,NZ0,NY0,NX0), wxp=gval<0,0,1>(w,z,y,x+1,NZ0,NY0,NX0);
  float wym=gval<0,0,1>(w,z,y-1,x,NZ0,NY0,NX0), wyp=gval<0,0,1>(w,z,y+1,x,NZ0,NY0,NX0);
  float wzm=gval<0,0,1>(w,z-1,y,x,NZ0,NY0,NX0), wzp=gval<0,0,1>(w,z+1,y,x,NZ0,NY0,NX0);
  float rhs_w = (wxm+wxp+wym+wyp+wzm+wzp - 6.0f*wc)*NU_DX2
              - (uc*(wxp-wxm) + vc*(wyp-wym) + wc*(wzp-wzm))*INV2DX;

  float a = al[i]; a = a < 0.05f ? 0.05f : (a > 1.0f ? 1.0f : a);
  float rho = a*RHOL_C + (1.0f - a)*RHOG_C;

  us[i]  = uc + DT_C*rhs_u;
  vs[i]  = vc + DT_C*rhs_v;
  wst[i] = wc + DT_C*rhs_w + DT_C*GZ_C*(rho - RHOL_C)/rho;
}

// ---------------- divergence -> Poisson RHS b ----------------
__global__ void div_b_kernel(const float* __restrict__ us, const float* __restrict__ vs,
                             const float* __restrict__ wst, float* __restrict__ bb){
  int i = blockIdx.x*blockDim.x + threadIdx.x;
  if (i >= NCELL) return;
  int x = i % NX0; int t = i / NX0; int y = t % NY0; int z = t / NY0;
  float d = (gval<1,0,0>(us,z,y,x+1,NZ0,NY0,NX0) - gval<1,0,0>(us,z,y,x-1,NZ0,NY0,NX0)
           + gval<0,1,0>(vs,z,y+1,x,NZ0,NY0,NX0) - gval<0,1,0>(vs,z,y-1,x,NZ0,NY0,NX0)
           + gval<0,0,1>(wst,z+1,y,x,NZ0,NY0,NX0) - gval<0,0,1>(wst,z-1,y,x,NZ0,NY0,NX0))
          * INV2DX;
  bb[i] = d * BSCALE;
}

// ---------------- level-0 A-operator tile kernel (WMMA) ----------------
// One wave owns a 16(y)x16(x) tile at fixed z. The x+center part of A·pd is a
// banded 20x16 matmul done with 5 chained V_WMMA_F32_16X16X4_F32; y/z neighbour
// terms via VALU. SMOOTH=1: out = pd - (A pd - b)/diag. SMOOTH=0: out = A pd - b.
template<int SMOOTH>
__global__ void applyA0_kernel(const float* __restrict__ pd, const float* __restrict__ bb,
                               float* __restrict__ out){
  const int lane = threadIdx.x & 31;
  const int wv   = threadIdx.x >> 5;
  const int tile = blockIdx.x*8 + wv;            // 8192 tiles = 64 z * 4 ty * 32 tx
  const int txi  = tile & 31;
  const int tyi  = (tile >> 5) & 3;
  const int z    = tile >> 7;
  const int x0   = txi*16, y0 = tyi*16;
  const int half = lane >> 4, lp = lane & 15;

  float xc[8];                                    // x+center contribution, D-layout rows
#if USE_WMMA
  {
    const float* __restrict__ arow = pd + ((size_t)z*NY0 + (y0 + lp))*NX0;
    v8f acc = {};
#pragma unroll
    for (int c = 0; c < 5; ++c){
      int k0  = 4*c + 2*half;                     // A lanes 0-15: K=k0,k0+1 ; 16-31: K=k0+2,k0+3
      int xi0 = x0 - 2 + k0;
      int xa0 = iclamp(xi0, 0, NX0-1);            // pd x is Neumann (clamp)
      int xa1 = iclamp(xi0 + 1, 0, NX0-1);
      v2f a;  a.x = arow[xa0]; a.y = arow[xa1];
      // banded stencil matrix S[i][n] = w(i-2-n): branchless two-select build
      int d0 = k0 - 2 - lp;
      int d1 = d0 + 1;
      float bx = (d0 == 1 || d0 == -1) ? INV_DX2 : 0.0f;
      bx       = (d0 == 0) ? DIAG_C : bx;
      float by = (d1 == 1 || d1 == -1) ? INV_DX2 : 0.0f;
      by       = (d1 == 0) ? DIAG_C : by;
      v2f bm; bm.x = bx; bm.y = by;
      acc = __builtin_amdgcn_wmma_f32_16x16x4_f32(false, a, false, bm,
                                                  (short)0, acc, false, false);
    }
#pragma unroll
    for (int r = 0; r < 8; ++r) xc[r] = acc[r];
  }
#else
#pragma unroll
  for (int r = 0; r < 8; ++r){
    int m = r + 8*half; int y = y0 + m; int x = x0 + lp;
    size_t i = ((size_t)z*NY0 + y)*NX0 + x;
    float pm = gval<0,0,2>(pd,z,y,x-1,NZ0,NY0,NX0);
    float pp = gval<0,0,2>(pd,z,y,x+1,NZ0,NY0,NX0);
    xc[r] = (pm + pp - 6.0f*pd[i])*INV_DX2;
  }
#endif

#pragma unroll
  for (int r = 0; r < 8; ++r){
    int m = r + 8*half;                           // D-matrix: vgpr r holds M=r / M=r+8
    int y = y0 + m; int x = x0 + lp;
    size_t i = ((size_t)z*NY0 + y)*NX0 + x;
    float yz = (gval<0,0,2>(pd,z,y-1,x,NZ0,NY0,NX0) + gval<0,0,2>(pd,z,y+1,x,NZ0,NY0,NX0)
              + gval<0,0,2>(pd,z-1,y,x,NZ0,NY0,NX0) + gval<0,0,2>(pd,z+1,y,x,NZ0,NY0,NX0))
             * INV_DX2;
    float Ap = xc[r] + yz;
    if (SMOOTH){
      out[i] = pd[i] - (Ap - bb[i]) * INVDIAG;
    } else {
      out[i] = Ap - bb[i];                        // residual
    }
  }
}

// ---------------- restriction: TDM stages a ghost-free fine tile into LDS ----------------
__global__ void restrict_kernel(const float* __restrict__ fine, float* __restrict__ coarse,
                                int fnz, int fny, int fnx, int cny, int cnx,
                                int TCY, int TCX){
  __shared__ float tile[2*8*128];                 // max: 2 z-planes * (2*4) y * (2*64) x
  const int cz = blockIdx.z;
  const int y0 = blockIdx.y*TCY;
  const int x0 = blockIdx.x*TCX;
  const int fz0 = 2*cz, fy0 = 2*y0, fx0 = 2*x0;
  const int tx = 2*TCX, ty = 2*TCY;

#if USE_TDM
  if ((threadIdx.x >> 5) == 0) {                  // one wave issues the DMA
    const float* src = fine + (((size_t)fz0)*fny + fy0)*fnx + fx0;
    unsigned long long ga = (unsigned long long)(uintptr_t)src;
    unsigned lds_off = (unsigned)(uintptr_t)(&tile[0]);   // flat LDS addr low 32b = LDS byte addr
    // D# group 0: count=1, lds_addr, global_addr[56:0], type=2
    u32x4 g0;
    g0.x = 1u;
    g0.y = lds_off;
    g0.z = (unsigned)(ga & 0xFFFFFFFFull);
    g0.w = (unsigned)((ga >> 32) & 0x01FFFFFFull) | (2u << 30);
    // D# group 1: data_size=4B, tensor dims/strides (elements), 3-D tile
    long long s0 = (long long)fnx;
    long long s1 = (long long)fnx * (long long)fny;
    i32x8 g1;
    g1[0] = (int)(2u << 16);                                        // data_size=2 (4 bytes)
    g1[1] = (int)(((unsigned)fnx & 0xFFFFu) << 16);                 // tensor_dim0 lo16
    g1[2] = (int)((((unsigned)fnx >> 16) & 0xFFFFu)
                | (((unsigned)fny & 0xFFFFu) << 16));               // dim0 hi | dim1 lo
    g1[3] = (int)((((unsigned)fny >> 16) & 0xFFFFu)
                | (((unsigned)tx  & 0xFFFFu) << 16));               // dim1 hi | tile_dim0
    g1[4] = (int)(((unsigned)ty & 0xFFFFu) | (2u << 16));           // tile_dim1 | tile_dim2=2
    g1[5] = (int)(unsigned)(s0 & 0xFFFFFFFFll);                     // stride0 lo32
    g1[6] = (int)(((unsigned)((s0 >> 32) & 0xFFFFll))
                | (((unsigned)(s1 & 0xFFFFll)) << 16));             // stride0 hi | stride1 lo
    g1[7] = (int)(unsigned)((s1 >> 16) & 0xFFFFFFFFll);             // stride1 hi32
    i32x4 g2; g2[0] = fnz; g2[1] = 0; g2[2] = 0; g2[3] = 0;         // tensor_dim2
    i32x4 g3; g3[0] = 0;   g3[1] = 0; g3[2] = 0; g3[3] = 0;
#if defined(__clang_major__) && (__clang_major__ >= 23)
    i32x8 g4; g4[0]=0; g4[1]=0; g4[2]=0; g4[3]=0; g4[4]=0; g4[5]=0; g4[6]=0; g4[7]=0;
    __builtin_amdgcn_tensor_load_to_lds(g0, g1, g2, g3, g4, 0);
#else
    __builtin_amdgcn_tensor_load_to_lds(g0, g1, g2, g3, 0);
#endif
#if __has_builtin(__builtin_amdgcn_s_wait_tensorcnt)
    __builtin_amdgcn_s_wait_tensorcnt(0);
#else
    asm volatile("s_wait_tensorcnt 0x0" ::: "memory");
#endif
  }
  __syncthreads();
#else
  for (int t = threadIdx.x; t < 2*ty*tx; t += blockDim.x){
    int fx = t % tx; int rem = t / tx; int fy = rem % ty; int fz = rem / ty;
    tile[t] = fine[(((size_t)(fz0+fz))*fny + (fy0+fy))*fnx + (fx0+fx)];
  }
  __syncthreads();
#endif

  if ((int)threadIdx.x < TCX*TCY){
    int cx = threadIdx.x % TCX;
    int cy = threadIdx.x / TCX;
    float s = 0.0f;
#pragma unroll
    for (int dz = 0; dz < 2; ++dz)
#pragma unroll
      for (int dy = 0; dy < 2; ++dy)
#pragma unroll
        for (int dx = 0; dx < 2; ++dx)
          s += tile[((dz*ty) + (2*cy+dy))*tx + (2*cx+dx)];
    coarse[((size_t)cz*cny + (y0+cy))*cnx + (x0+cx)] = 0.125f*s;
  }
}

// ---------------- up-sweep: wt = P(coarse) - (A P(coarse) - r)/diag (fused prolong) ----------------
__device__ __forceinline__ float pval(const float* __restrict__ c, int z,int y,int x,
                                      int nz,int ny,int nx,int cny,int cnx){
  bool zero = (z >= nz);                          // pd top-z Dirichlet at this level
  x = iclamp(x, 0, nx-1);
  y = iclamp(y, 0, ny-1);
  z = iclamp(z, 0, nz-1);
  float v = c[((size_t)(z>>1)*cny + (y>>1))*cnx + (x>>1)];
  return zero ? 0.0f : v;
}

__global__ void smooth_up_kernel(const float* __restrict__ coarse, const float* __restrict__ r,
                                 float* __restrict__ out, int nz,int ny,int nx,
                                 int cny,int cnx){
  int n = nz*ny*nx;
  int i = blockIdx.x*blockDim.x + threadIdx.x;
  if (i >= n) return;
  if (coarse == nullptr){ out[i] = r[i]*INVDIAG; return; }   // w=0 start at coarsest level
  int x = i % nx; int t = i / nx; int y = t % ny; int z = t / ny;
  float c = pval(coarse,z,y,x,nz,ny,nx,cny,cnx);
  float s = pval(coarse,z,y,x-1,nz,ny,nx,cny,cnx) + pval(coarse,z,y,x+1,nz,ny,nx,cny,cnx)
          + pval(coarse,z,y-1,x,nz,ny,nx,cny,cnx) + pval(coarse,z,y+1,x,nz,ny,nx,cny,cnx)
          + pval(coarse,z-1,y,x,nz,ny,nx,cny,cnx) + pval(coarse,z+1,y,x,nz,ny,nx,cny,cnx);
  float Ap = (s - 6.0f*c)*INV_DX2;
  out[i] = c - (Ap - r[i])*INVDIAG;
}

// ---------------- level 0: pd_out = pd_in - prolong(wt1) ----------------
__global__ void prolong_sub_kernel(const float* __restrict__ pdin, const float* __restrict__ cw,
                                   float* __restrict__ pdout){
  int i = blockIdx.x*blockDim.x + threadIdx.x;
  if (i >= NCELL) return;
  int x = i % NX0; int t = i / NX0; int y = t % NY0; int z = t / NY0;
  pdout[i] = pdin[i] - cw[((size_t)(z>>1)*32 + (y>>1))*256 + (x>>1)];
}

// ---------------- projection + output assembly ----------------
__global__ void project_kernel(const float* __restrict__ al, const float* __restrict__ us,
                               const float* __restrict__ vs, const float* __restrict__ wst,
                               const float* __restrict__ pd, float* __restrict__ out){
  int i = blockIdx.x*blockDim.x + threadIdx.x;
  if (i >= NCELL) return;
  int x = i % NX0; int t = i / NX0; int y = t % NY0; int z = t / NY0;
  float dpx = (gval<0,0,2>(pd,z,y,x+1,NZ0,NY0,NX0) - gval<0,0,2>(pd,z,y,x-1,NZ0,NY0,NX0))*INV2DX;
  float dpy = (gval<0,0,2>(pd,z,y+1,x,NZ0,NY0,NX0) - gval<0,0,2>(pd,z,y-1,x,NZ0,NY0,NX0))*INV2DX;
  float dpz = (gval<0,0,2>(pd,z+1,y,x,NZ0,NY0,NX0) - gval<0,0,2>(pd,z-1,y,x,NZ0,NY0,NX0))*INV2DX;
  float a = al[i]; a = a < 0.05f ? 0.05f : (a > 1.0f ? 1.0f : a);
  float rho = a*RHOL_C + (1.0f - a)*RHOG_C;
  float cf = DT_C / rho;
  out[0*(size_t)NCELL + i] = us[i]  - dpx*cf;
  out[1*(size_t)NCELL + i] = vs[i]  - dpy*cf;
  out[2*(size_t)NCELL + i] = wst[i] - dpz*cf;
  out[3*(size_t)NCELL + i] = pd[i];
}

// ---------------- host driver ----------------
extern "C" void kernel_launch(void* const* d_in, const int* in_sizes, int n_in,
                              void* d_out, int out_size, void* d_ws, size_t ws_size,
                              hipStream_t stream){
  (void)in_sizes; (void)n_in; (void)out_size;
  const float* alpha = (const float*)d_in[0];
  const float* u     = (const float*)d_in[1];
  const float* v     = (const float*)d_in[2];
  const float* w     = (const float*)d_in[3];
  const float* pd0   = (const float*)d_in[4];
  float* out = (float*)d_out;

  static const int LZv[7] = {64,32,16,8,4,2,1};
  static const int LYv[7] = {64,32,16,8,4,2,1};
  static const int LXv[7] = {512,256,128,64,32,16,8};
  int lsz[7];
  size_t tot = 0;
  for (int l = 0; l < 7; ++l){ lsz[l] = LZv[l]*LYv[l]*LXv[l]; tot += (size_t)lsz[l]; }
  size_t need = (6*(size_t)NCELL + tot + (tot - (size_t)lsz[0])) * sizeof(float);
  if (ws_size < need) return;                     // scratch too small: bail safely

  float* f   = (float*)d_ws;
  float* us  = f; f += NCELL;
  float* vs  = f; f += NCELL;
  float* wst = f; f += NCELL;
  float* bb  = f; f += NCELL;
  float* pda = f; f += NCELL;
  float* pdb = f; f += NCELL;
  float* r[7];  for (int l = 0; l < 7; ++l){ r[l] = f; f += lsz[l]; }
  float* wt[7]; wt[0] = nullptr;
  for (int l = 1; l < 7; ++l){ wt[l] = f; f += lsz[l]; }

  const int nb0 = (NCELL + 255)/256;

  predictor_kernel<<<nb0, 256, 0, stream>>>(alpha, u, v, w, us, vs, wst);
  div_b_kernel   <<<nb0, 256, 0, stream>>>(us, vs, wst, bb);

  const float* pin = pd0;
  float* pcur = pda; float* palt = pdb; float* final_pd = pda;
  for (int cyc = 0; cyc < 2; ++cyc){
    applyA0_kernel<1><<<1024, 256, 0, stream>>>(pin, bb, pcur);               // pre-smooth
    applyA0_kernel<0><<<1024, 256, 0, stream>>>(pcur, bb, r[0]);              // fine residual
    for (int l = 0; l < 6; ++l){                                              // restrict chain
      int cnz = LZv[l+1], cny = LYv[l+1], cnx = LXv[l+1];
      int TCX = cnx < 64 ? cnx : 64;
      int TCY = cny < 4  ? cny : 4;
      dim3 grid(cnx/TCX, cny/TCY, cnz);
      restrict_kernel<<<grid, 256, 0, stream>>>(r[l], r[l+1],
                                                LZv[l], LYv[l], LXv[l], cny, cnx, TCY, TCX);
    }
    smooth_up_kernel<<<(lsz[6]+255)/256, 256, 0, stream>>>(nullptr, r[6], wt[6],
                                                           LZv[6], LYv[6], LXv[6], 1, 1);
    for (int l = 5; l >= 1; --l)                                              // fused prolong+smooth
      smooth_up_kernel<<<(lsz[l]+255)/256, 256, 0, stream>>>(wt[l+1], r[l], wt[l],
                                                             LZv[l], LYv[l], LXv[l],
                                                             LYv[l+1], LXv[l+1]);
    prolong_sub_kernel<<<nb0, 256, 0, stream>>>(pcur, wt[1], palt);           // pd -= P(w)
    applyA0_kernel<1><<<1024, 256, 0, stream>>>(palt, bb, pcur);              // post-smooth
    final_pd = pcur;
    pin = final_pd;
    float* tmp = pcur; pcur = palt; palt = tmp;
  }

  project_kernel<<<nb0, 256, 0, stream>>>(alpha, us, vs, wst, final_pd, out);
}